// LinearAttentionSimilarity_75617194213721
// MI455X (gfx1250) — compile-verified
//
#include <hip/hip_runtime.h>

typedef __attribute__((ext_vector_type(2))) float v2f;
typedef __attribute__((ext_vector_type(8))) float v8f;

#define EPSF 1e-6f

constexpr int N_ = 8;
constexpr int L_ = 2048;
constexpr int S_ = 2048;
constexpr int H_ = 16;
constexpr int D_ = 64;
constexpr int CH_ = 8;          // L-chunks per head in pass 2
constexpr int NH_ = N_ * H_;    // 128 heads

__device__ __forceinline__ float phi_elu1(float x) {
  // elu(x) + 1  ==  x>0 ? x+1 : exp(x)
  return x > 0.f ? x + 1.f : __expf(x);
}

__device__ __forceinline__ v8f wmma_f32(v2f a, v2f b, v8f c) {
  // D = A(16x4, f32) * B(4x16, f32) + C(16x16, f32)
  return __builtin_amdgcn_wmma_f32_16x16x4_f32(false, a, false, b, (short)0, c,
                                               false, false);
}

// ---------------------------------------------------------------------------
// Pass 1: per head, KK = K^T K (64x64) and K_sum (64). One workgroup per head,
// 8 waves each own 256 rows of S; deterministic wave-serialized LDS reduce.
// ---------------------------------------------------------------------------
__global__ __launch_bounds__(256) void la_pass1(
    const float* __restrict__ keys, const float* __restrict__ key_mask,
    float* __restrict__ kk_out, float* __restrict__ ksum_out) {
  __shared__ float kk_lds[64][80];      // stride 80 -> bank spread
  __shared__ float ksum_lds[64][2];

  const int head = blockIdx.x;          // 0..127
  const int n = head / H_;
  const int h = head % H_;
  const int tid = threadIdx.x;
  const int wave = tid >> 5;
  const int lane = tid & 31;
  const int lh = lane >> 4;             // half-wave index
  const int lm = lane & 15;

  for (int i = tid; i < 64 * 80; i += 256) (&kk_lds[0][0])[i] = 0.f;
  if (tid < 128) (&ksum_lds[0][0])[tid] = 0.f;
  __syncthreads();

  const v8f vz = {0.f, 0.f, 0.f, 0.f, 0.f, 0.f, 0.f, 0.f};
  v8f acc[4][4];
#pragma unroll
  for (int i = 0; i < 4; ++i)
#pragma unroll
    for (int j = 0; j < 4; ++j) acc[i][j] = vz;
  float ksum_part[4] = {0.f, 0.f, 0.f, 0.f};

  const int rows_per_wave = S_ / 8;     // 256
  const int s_begin = wave * rows_per_wave;
  for (int s0 = s_begin; s0 < s_begin + rows_per_wave; s0 += 4) {
    const int r0 = s0 + 2 * lh;         // this lane's v=0 row; v=1 row is r0+1
    const float km0 = key_mask[n * S_ + r0];
    const float km1 = key_mask[n * S_ + r0 + 1];
    const float* p0 = keys + (size_t)((n * S_ + r0) * H_ + h) * D_;
    const float* p1 = p0 + (size_t)H_ * D_;
    v2f frag[4];                        // A-frag == B-frag for K^T K
#pragma unroll
    for (int i = 0; i < 4; ++i) {
      float a0 = phi_elu1(p0[16 * i + lm]) * km0;
      float a1 = phi_elu1(p1[16 * i + lm]) * km1;
      frag[i].x = a0;
      frag[i].y = a1;
      ksum_part[i] += a0 + a1;          // column 16*i+lm partial sum
    }
#pragma unroll
    for (int i = 0; i < 4; ++i)
#pragma unroll
      for (int j = 0; j < 4; ++j)
        acc[i][j] = wmma_f32(frag[i], frag[j], acc[i][j]);
  }

  // deterministic cross-wave accumulation (one wave at a time)
  for (int w = 0; w < 8; ++w) {
    if (wave == w) {
#pragma unroll
      for (int i = 0; i < 4; ++i) {
#pragma unroll
        for (int j = 0; j < 4; ++j) {
#pragma unroll
          for (int r = 0; r < 8; ++r)
            kk_lds[16 * i + r + 8 * lh][16 * j + lm] += acc[i][j][r];
        }
        ksum_lds[16 * i + lm][lh] += ksum_part[i];
      }
    }
    __syncthreads();
  }

  float* kkdst = kk_out + head * (D_ * D_);
  for (int i = tid; i < D_ * D_; i += 256) kkdst[i] = kk_lds[i >> 6][i & 63];
  if (tid < 64) ksum_out[head * 64 + tid] = ksum_lds[tid][0] + ksum_lds[tid][1];
}

// ---------------------------------------------------------------------------
// Pass 2: per (head, L-chunk): tmp = phi(Q)@KK via WMMA, row dots with phi(Q),
// accumulate sum_l max(q'KKq,eps)/max((q.Ksum+eps)^2,eps) * qmask.
// ---------------------------------------------------------------------------
__global__ __launch_bounds__(256) void la_pass2(
    const float* __restrict__ queries, const float* __restrict__ query_mask,
    const float* __restrict__ kk_in, const float* __restrict__ ksum_in,
    float* __restrict__ partial) {
  __shared__ float kk_lds[64][80];
  __shared__ float ksum_lds[64];
  __shared__ float q_lds[8][16][68];    // per-wave phi(Q) tile, stride 68
  __shared__ float red[8][16][16];
  __shared__ float red2[8][16][2];
  __shared__ float redf[8][16];

  const int head = blockIdx.x / CH_;
  const int chunk = blockIdx.x % CH_;
  const int n = head / H_;
  const int h = head % H_;
  const int tid = threadIdx.x;
  const int wave = tid >> 5;
  const int lane = tid & 31;
  const int lh = lane >> 4;
  const int lm = lane & 15;

  for (int i = tid; i < D_ * D_; i += 256)
    kk_lds[i >> 6][i & 63] = kk_in[head * (D_ * D_) + i];
  if (tid < 64) ksum_lds[tid] = ksum_in[head * 64 + tid];
  __syncthreads();

  const int lchunk = L_ / CH_;                       // 256 rows per block
  const int l_wave = chunk * lchunk + wave * (lchunk / 8);  // 32 rows per wave
  const v8f vz = {0.f, 0.f, 0.f, 0.f, 0.f, 0.f, 0.f, 0.f};

  float wacc = 0.f;
  for (int t = 0; t < 2; ++t) {                      // two 16-row tiles / wave
    const int l0 = l_wave + t * 16;
    // stage phi(Q) tile (coalesced 128B rows); same-wave LDS is in-order
#pragma unroll
    for (int k = 0; k < 32; ++k) {
      int row = k >> 1;
      int col = ((k & 1) << 5) + lane;
      float x = queries[(size_t)((n * L_ + l0 + row) * H_ + h) * D_ + col];
      q_lds[wave][row][col] = phi_elu1(x);
    }
    // tmp(16x64) = Qtile(16x64) @ KK(64x64): 16 k-chunks x 4 n-blocks
    v8f acc[4] = {vz, vz, vz, vz};
#pragma unroll
    for (int c = 0; c < 16; ++c) {
      v2f a;
      a.x = q_lds[wave][lm][4 * c + 2 * lh];
      a.y = q_lds[wave][lm][4 * c + 2 * lh + 1];
#pragma unroll
      for (int mb = 0; mb < 4; ++mb) {
        v2f b;
        b.x = kk_lds[4 * c + 2 * lh][16 * mb + lm];
        b.y = kk_lds[4 * c + 2 * lh + 1][16 * mb + lm];
        acc[mb] = wmma_f32(a, b, acc[mb]);
      }
    }
    // per-lane partial of sum_m tmp[l,m]*phiQ[l,m]
#pragma unroll
    for (int r = 0; r < 8; ++r) {
      float s = 0.f;
#pragma unroll
      for (int mb = 0; mb < 4; ++mb)
        s += acc[mb][r] * q_lds[wave][r + 8 * lh][16 * mb + lm];
      red[wave][r + 8 * lh][lm] = s;
    }
    // per-lane partial of q . Ksum  (2 lanes per row, 32 d each)
    float s2 = 0.f;
#pragma unroll
    for (int k = 0; k < 32; ++k)
      s2 += q_lds[wave][lm][32 * lh + k] * ksum_lds[32 * lh + k];
    red2[wave][lm][lh] = s2;
    // finish rows (same-wave DS ops are in-order; fixed summation order)
    if (lane < 16) {
      float num = 0.f;
#pragma unroll
      for (int c2 = 0; c2 < 16; ++c2) num += red[wave][lm][c2];
      float sden = red2[wave][lm][0] + red2[wave][lm][1] + EPSF;
      float den = fmaxf(sden * sden, EPSF);
      float qm = query_mask[n * L_ + l0 + lm];
      wacc += fmaxf(num, EPSF) / den * qm;
    }
  }
  if (lane < 16) redf[wave][lm] = wacc;
  __syncthreads();
  if (tid == 0) {
    float s = 0.f;
    for (int i = 0; i < 128; ++i) s += (&redf[0][0])[i];
    partial[head * CH_ + chunk] = s;
  }
}

// ---------------------------------------------------------------------------
// Pass 3: counts from masks, tau per head, mean over heads -> out[N].
// ---------------------------------------------------------------------------
__global__ __launch_bounds__(256) void la_pass3(
    const float* __restrict__ query_mask, const float* __restrict__ key_mask,
    const float* __restrict__ partial, float* __restrict__ out) {
  __shared__ float redq[256];
  __shared__ float redk[256];
  __shared__ float qc[N_], kc[N_];
  __shared__ float tau[NH_];
  const int tid = threadIdx.x;

  for (int nn = 0; nn < N_; ++nn) {
    float sq = 0.f, sk = 0.f;
    for (int i = tid; i < L_; i += 256) sq += query_mask[nn * L_ + i];
    for (int i = tid; i < S_; i += 256) sk += key_mask[nn * S_ + i];
    redq[tid] = sq;
    redk[tid] = sk;
    __syncthreads();
    for (int off = 128; off > 0; off >>= 1) {
      if (tid < off) {
        redq[tid] += redq[tid + off];
        redk[tid] += redk[tid + off];
      }
      __syncthreads();
    }
    if (tid == 0) {
      qc[nn] = fmaxf(redq[0], 1.f);
      kc[nn] = fmaxf(redk[0], 1.f);
    }
    __syncthreads();
  }

  if (tid < NH_) {
    int nn = tid / H_;
    float s = 0.f;
    for (int c = 0; c < CH_; ++c) s += partial[tid * CH_ + c];
    float mean = s / qc[nn];
    float kcn = kc[nn];
    float tv = (kcn * mean - 1.f) / fmaxf(kcn - 1.f, 1e-6f);
    tau[tid] = fminf(fmaxf(tv, 0.f), 1.f);
  }
  __syncthreads();
  if (tid < N_) {
    float s = 0.f;
    for (int hh = 0; hh < H_; ++hh) s += tau[tid * H_ + hh];
    out[tid] = s / (float)H_;
  }
}

// ---------------------------------------------------------------------------
extern "C" void kernel_launch(void* const* d_in, const int* in_sizes, int n_in,
                              void* d_out, int out_size, void* d_ws,
                              size_t ws_size, hipStream_t stream) {
  const float* queries = (const float*)d_in[0];
  const float* keys = (const float*)d_in[1];
  // d_in[2] (values) is unused by the reference op
  const float* query_mask = (const float*)d_in[3];
  const float* key_mask = (const float*)d_in[4];
  float* out = (float*)d_out;

  float* kk = (float*)d_ws;                 // NH_ * 4096 floats (2 MB)
  float* ksum = kk + (size_t)NH_ * D_ * D_; // NH_ * 64
  float* partial = ksum + (size_t)NH_ * D_; // NH_ * CH_

  la_pass1<<<NH_, 256, 0, stream>>>(keys, key_mask, kk, ksum);
  la_pass2<<<NH_ * CH_, 256, 0, stream>>>(queries, query_mask, kk, ksum,
                                          partial);
  la_pass3<<<1, 256, 0, stream>>>(query_mask, key_mask, partial, out);
}